// TradingModule_37924561224004
// MI455X (gfx1250) — compile-verified
//
#include <hip/hip_runtime.h>
#include <hip/hip_bf16.h>

typedef float v2f __attribute__((ext_vector_type(2)));
typedef float v8f __attribute__((ext_vector_type(8)));

#define STARTING_CASH 10000.0f

// ---------------------------------------------------------------------------
// Kernel 1: single wave32. action = tanh(b1) . W2 + b2, computed in fp32 on
// the matrix pipe via V_WMMA_F32_16X16X4_F32.
//
//   A[m][k] = tanh(b1[64*ch + 4m + k])   (16x4, fp32)
//   B[k][n] =       W2[64*ch + 4n + k]   (4x16, fp32)
//   D += A x B  (two chained WMMAs cover NQ=128)
//   dot = trace(D)  -> action -> buy = trunc(max(0,action))
//   shares_final = timesteps * buy  (int32 wrap semantics, like jnp)
//
// A layout (ISA 7.12.2, 32-bit 16x4): lanes 0-15 hold K=0(V0),1(V1) for M=lane;
// lanes 16-31 hold K=2,3 for M=lane-16. B uses the same half-wave K split with
// N = lane%16 — so every lane loads A and B at the SAME flat indices.
// ---------------------------------------------------------------------------
__global__ __launch_bounds__(32)
void trade_action_kernel(const float* __restrict__ b1,
                         const float* __restrict__ W2,
                         const float* __restrict__ b2,
                         int nq, int timesteps,
                         int* __restrict__ ws_shares)
{
    const int lane = threadIdx.x;            // 0..31, EXEC all-ones (no divergence)
    const int l16  = lane & 15;
    const int klo  = (lane < 16) ? 0 : 2;

    v8f c = {};                              // fp32 accumulator, chained D->C

    const int nchunks = (nq + 63) / 64;      // 2 for NQ=128
    for (int ch = 0; ch < nchunks; ++ch) {
        const int j0 = ch * 64 + 4 * l16 + klo;
        const int j1 = j0 + 1;
        const float h0 = (j0 < nq) ? tanhf(b1[j0]) : 0.0f;
        const float h1 = (j1 < nq) ? tanhf(b1[j1]) : 0.0f;
        const float w0 = (j0 < nq) ? W2[j0] : 0.0f;
        const float w1 = (j1 < nq) ? W2[j1] : 0.0f;
        v2f a = { h0, h1 };
        v2f b = { w0, w1 };
        // 8 args: (neg_a, A, neg_b, B, c_mod, C, reuse_a, reuse_b)
        c = __builtin_amdgcn_wmma_f32_16x16x4_f32(
                false, a, false, b, (short)0, c, false, false);
    }

    // Trace of D. C/D layout (16x16 f32): lanes 0-15 VGPR r -> D[r][lane];
    // lanes 16-31 VGPR r -> D[r+8][lane-16].
    //   D[m][m], m<8 : lane m,    c[m]
    //   D[m][m], m>=8: lane m+16, c[m-8]  (lanes 24..31 -> c[lane-24])
    __shared__ float red[32];
    float contrib = 0.0f;
    if (lane < 8)        contrib = c[lane];
    else if (lane >= 24) contrib = c[lane - 24];
    red[lane] = contrib;
    __syncthreads();

    if (lane == 0) {
        float dot = 0.0f;
        for (int i = 0; i < 32; ++i) dot += red[i];
        const float action = dot + b2[0];
        // curprice == 0  =>  cash/0 = +inf (cash stays 10000 > 0 forever),
        // buy = trunc(max(0, action)) every step; sell = 0 in all cases.
        const int buy = (int)fmaxf(0.0f, action);
        const unsigned shares = (unsigned)timesteps * (unsigned)buy; // i32 wrap
        ws_shares[0] = (int)shares;
    }
}

// ---------------------------------------------------------------------------
// Kernel 2: out[i] = 10000 + shares_final * quotes[i, qlen-1]
// 8192 strided loads, one cacheline each (~1 MB effective HBM traffic).
// Separate mul then add (round-to-nearest each) to mirror the reference's
// `cash + shares * price` without fused contraction.
// ---------------------------------------------------------------------------
__global__ void trade_payout_kernel(const float* __restrict__ quotes,
                                    const int* __restrict__ ws_shares,
                                    float* __restrict__ out,
                                    int batch, int qlen)
{
    const int i = blockIdx.x * blockDim.x + threadIdx.x;
    if (i >= batch) return;
    const float price  = quotes[(size_t)i * (size_t)qlen + (size_t)(qlen - 1)];
    const float shares = (float)ws_shares[0];   // uniform -> scalar load
    out[i] = __fadd_rn(STARTING_CASH, __fmul_rn(shares, price));
}

extern "C" void kernel_launch(void* const* d_in, const int* in_sizes, int n_in,
                              void* d_out, int out_size, void* d_ws, size_t ws_size,
                              hipStream_t stream)
{
    const float* all_quotes = (const float*)d_in[0];
    // d_in[1] = W1: provably dead — `inputs` is all zeros, so inputs @ W1.T == 0.
    const float* b1 = (const float*)d_in[2];
    const float* W2 = (const float*)d_in[3];
    const float* b2 = (const float*)d_in[4];

    const int nq        = in_sizes[2];          // 128
    const int batch     = out_size;             // 8192
    const int qlen      = in_sizes[0] / batch;  // 4224
    const int timesteps = qlen - nq;            // 4096

    int* ws_shares = (int*)d_ws;

    trade_action_kernel<<<1, 32, 0, stream>>>(b1, W2, b2, nq, timesteps, ws_shares);

    const int tpb = 256;                        // 8 wave32s per block
    trade_payout_kernel<<<(batch + tpb - 1) / tpb, tpb, 0, stream>>>(
        all_quotes, ws_shares, (float*)d_out, batch, qlen);
}